// Attention_28724741275862
// MI455X (gfx1250) — compile-verified
//
#include <hip/hip_runtime.h>

typedef __attribute__((ext_vector_type(2))) float v2f;
typedef __attribute__((ext_vector_type(8))) float v8f;

#define H   512   // hidden (H2)
#define TQ  64    // decoder length
#define TK  512   // encoder length
#define NB  8     // batch

// ---------------------------------------------------------------------------
// C[m,n] = sum_k A[m,k] * W[n, koff+k]  (+ bias[n])   K=H, lda=H, ldw=2H, ldc=H
// One wave (32 lanes) per 16x16 C tile; V_WMMA_F32_16X16X4_F32 over K.
// A-frag (16x4 f32): lane L -> M=L&15, values A[M][k+2*(L>>4) + {0,1}]
// B-frag (4x16 f32): lane L -> N=L&15, values B[k+2*(L>>4)+{0,1}][N]
// C/D:               lane L -> N=L&15, VGPR v -> M = v + 8*(L>>4)
// ---------------------------------------------------------------------------
__global__ __launch_bounds__(32)
void gemm_awt_bias(const float* __restrict__ A, const float* __restrict__ W,
                   const float* __restrict__ bias, float* __restrict__ C, int koff) {
  const int nt = blockIdx.x, mt = blockIdx.y;
  const int lane = threadIdx.x;
  const int half = lane >> 4, lm = lane & 15;
  const float* arow = A + (mt * 16 + lm) * H;
  const float* wrow = W + (nt * 16 + lm) * (2 * H) + koff;
  v8f acc = {};
#pragma unroll 8
  for (int k = 0; k < H; k += 4) {
    v2f a = *(const v2f*)(arow + k + 2 * half);
    v2f b = *(const v2f*)(wrow + k + 2 * half);
    acc = __builtin_amdgcn_wmma_f32_16x16x4_f32(false, a, false, b,
                                                (short)0, acc, false, false);
  }
  const int n  = nt * 16 + lm;
  const float bb = bias ? bias[n] : 0.0f;
#pragma unroll
  for (int v = 0; v < 8; ++v) {
    const int m = mt * 16 + v + 8 * half;
    C[m * H + n] = acc[v] + bb;
  }
}

// ---------------------------------------------------------------------------
// S1[b,d] = sum_j enc[b,j,d] ; S2[b,d] = sum_j E[b,j,d]*enc[b,j,d]
// One thread per (b,d); loads coalesced across d.
// ---------------------------------------------------------------------------
__global__ __launch_bounds__(256)
void rowsums(const float* __restrict__ enc, const float* __restrict__ E,
             float* __restrict__ S1, float* __restrict__ S2) {
  const int idx = blockIdx.x * 256 + threadIdx.x;   // b*H + d
  const int b = idx / H;
  const int d = idx - b * H;
  const float* ep = enc + b * TK * H + d;
  const float* Ep = E   + b * TK * H + d;
  float s1 = 0.f, s2 = 0.f;
  for (int j = 0; j < TK; ++j) {
    const float x = ep[j * H];
    const float y = Ep[j * H];
    s1 += x;
    s2 += x * y;
  }
  S1[idx] = s1;
  S2[idx] = s2;
}

// ---------------------------------------------------------------------------
// LSE[b,i,j] = logsumexp_o( D[b,i,o] + E[b,j,o] )   (o = 0..511)
// One block per (b,i); 8 waves; each wave owns a contiguous range of j.
// Each lane covers 16 consecutive o; wave32 xor-shuffle reduction.
// ---------------------------------------------------------------------------
__global__ __launch_bounds__(256)
void lse_rows(const float* __restrict__ D, const float* __restrict__ E,
              float* __restrict__ LSE) {
  const int row  = blockIdx.x;          // b*TQ + i
  const int b    = row / TQ;
  const int lane = threadIdx.x & 31;
  const int wave = threadIdx.x >> 5;    // 0..7

  // D row (with bias already folded in) cached in registers: 16 floats/lane.
  const float* dptr = D + row * H + lane * 16;
  float dv[16];
  ((float4*)dv)[0] = ((const float4*)dptr)[0];
  ((float4*)dv)[1] = ((const float4*)dptr)[1];
  ((float4*)dv)[2] = ((const float4*)dptr)[2];
  ((float4*)dv)[3] = ((const float4*)dptr)[3];

  const int jbase = wave * (TK / 8);    // contiguous E rows per wave
  for (int t = 0; t < TK / 8; ++t) {
    const int j = jbase + t;
    const float* eptr = E + (b * TK + j) * H + lane * 16;
    __builtin_prefetch(eptr + H, 0, 0); // global_prefetch_b8 for next E row

    float ev[16];
    ((float4*)ev)[0] = ((const float4*)eptr)[0];
    ((float4*)ev)[1] = ((const float4*)eptr)[1];
    ((float4*)ev)[2] = ((const float4*)eptr)[2];
    ((float4*)ev)[3] = ((const float4*)eptr)[3];

    float vv[16];
    float m = -3.402823466e+38f;
#pragma unroll
    for (int k = 0; k < 16; ++k) {
      vv[k] = dv[k] + ev[k];
      m = fmaxf(m, vv[k]);
    }
#pragma unroll
    for (int off = 16; off; off >>= 1)
      m = fmaxf(m, __shfl_xor(m, off, 32));

    float s = 0.f;
#pragma unroll
    for (int k = 0; k < 16; ++k)
      s += __expf(vv[k] - m);
#pragma unroll
    for (int off = 16; off; off >>= 1)
      s += __shfl_xor(s, off, 32);

    if (lane == 0)
      LSE[row * TK + j] = m + __logf(s);
  }
}

// ---------------------------------------------------------------------------
// out[b,i,d] = D[b,i,d]*S1[b,d] + S2[b,d] - sum_j LSE[b,i,j]*enc[b,j,d]
// Batched WMMA GEMM (LSE 64x512 @ enc 512x512) fused with the rank-1 terms.
// ---------------------------------------------------------------------------
__global__ __launch_bounds__(32)
void finalize(const float* __restrict__ LSE, const float* __restrict__ enc,
              const float* __restrict__ D,   const float* __restrict__ S1,
              const float* __restrict__ S2,  float* __restrict__ out) {
  const int nt = blockIdx.x, mt = blockIdx.y, b = blockIdx.z;
  const int lane = threadIdx.x;
  const int half = lane >> 4, lm = lane & 15;
  const float* Ab = LSE + b * TQ * TK + (mt * 16 + lm) * TK;
  const float* Bb = enc + b * TK * H;
  const int n = nt * 16 + lm;
  v8f acc = {};
#pragma unroll 4
  for (int k = 0; k < TK; k += 4) {
    v2f a = *(const v2f*)(Ab + k + 2 * half);
    const int kk = k + 2 * half;
    v2f bf;
    bf.x = Bb[kk * H + n];         // enc is row-major (j,d): rows kk, kk+1
    bf.y = Bb[(kk + 1) * H + n];
    acc = __builtin_amdgcn_wmma_f32_16x16x4_f32(false, a, false, bf,
                                                (short)0, acc, false, false);
  }
  const float s1 = S1[b * H + n];
  const float s2 = S2[b * H + n];
#pragma unroll
  for (int v = 0; v < 8; ++v) {
    const int i = mt * 16 + v + 8 * half;
    const float dd = D[(b * TQ + i) * H + n];
    out[(b * TQ + i) * H + n] = dd * s1 + s2 - acc[v];
  }
}

// ---------------------------------------------------------------------------
extern "C" void kernel_launch(void* const* d_in, const int* in_sizes, int n_in,
                              void* d_out, int out_size, void* d_ws, size_t ws_size,
                              hipStream_t stream) {
  (void)in_sizes; (void)n_in; (void)out_size; (void)ws_size;
  const float* enc  = (const float*)d_in[0];  // (8,512,512)
  const float* dec  = (const float*)d_in[1];  // (8,64,512)
  const float* W    = (const float*)d_in[2];  // (512,1024)
  const float* bias = (const float*)d_in[3];  // (512,)
  float* out = (float*)d_out;                 // (8,64,512)

  float* ws   = (float*)d_ws;
  float* Dp   = ws;                                  // 8*64*512   = 262144
  float* Ep   = Dp   + NB * TQ * H;                  // 8*512*512  = 2097152
  float* LSEp = Ep   + NB * TK * H;                  // 8*64*512   = 262144
  float* S1p  = LSEp + NB * TQ * TK;                 // 8*512      = 4096
  float* S2p  = S1p  + NB * H;                       // 8*512      = 4096
  // total ~10.5 MB of workspace

  // D = dec @ W_dec^T + bias   (M = B*TQ = 512)
  gemm_awt_bias<<<dim3(H / 16, (NB * TQ) / 16), 32, 0, stream>>>(dec, W, bias, Dp, 0);
  // E = enc @ W_enc^T          (M = B*TK = 4096)
  gemm_awt_bias<<<dim3(H / 16, (NB * TK) / 16), 32, 0, stream>>>(enc, W, nullptr, Ep, H);
  // S1, S2
  rowsums<<<(NB * H) / 256, 256, 0, stream>>>(enc, Ep, S1p, S2p);
  // LSE (the dominant exp phase): one block per (b,i)
  lse_rows<<<NB * TQ, 256, 0, stream>>>(Dp, Ep, LSEp);
  // out = D*S1 + S2 - LSE @ enc
  finalize<<<dim3(H / 16, TQ / 16, NB), 32, 0, stream>>>(LSEp, enc, Dp, S1p, S2p, out);
}